// BiLSTM_20985210208614
// MI455X (gfx1250) — compile-verified
//
#include <hip/hip_runtime.h>
#include <hip/hip_bf16.h>
#include <math.h>

#define H 128
#define T 16384
#define LAYERS 5

typedef __attribute__((ext_vector_type(2))) float v2f;
typedef __attribute__((ext_vector_type(8))) float v8f;

// Fast device nonlinearities: v_exp_f32 + v_rcp_f32, avoids ocml tanh/div.
__device__ __forceinline__ float sigm(float x) {
    return __builtin_amdgcn_rcpf(1.0f + __expf(-x));
}
__device__ __forceinline__ float tanh_fast(float x) {
    // tanh(x) = 2*sigmoid(2x) - 1 ; saturates correctly for large |x|
    return 2.0f * sigm(2.0f * x) - 1.0f;
}

// ---------------------------------------------------------------------------
// Layer-0 gate precompute: input width 6, trivially parallel VALU kernel.
// pre[dir][t][j] = sum_k x[t][k]*W0[dir][j][k] + b_ih[0,dir,j] + b_hh[0,dir,j]
// ---------------------------------------------------------------------------
__global__ void lstm_pre_l0(const float* __restrict__ x,      // [T][6]
                            const float* __restrict__ w0,     // [2][512][6]
                            const float* __restrict__ bih,    // [2][512] (layer 0 base)
                            const float* __restrict__ bhh,    // [2][512]
                            float* __restrict__ pre)          // [2][T][512]
{
    int idx = blockIdx.x * blockDim.x + threadIdx.x;      // 2*T*512 threads
    int j   = idx & 511;
    int t   = (idx >> 9) & (T - 1);
    int dir = idx >> 23;                                   // 2*T*512 = 2^24
    const float* xr = x + (size_t)t * 6;
    const float* wr = w0 + ((size_t)dir * 512 + j) * 6;
    float s = bih[dir * 512 + j] + bhh[dir * 512 + j];
    #pragma unroll
    for (int k = 0; k < 6; ++k) s += xr[k] * wr[k];
    pre[((size_t)dir * T + t) * 512 + j] = s;
}

// ---------------------------------------------------------------------------
// Layers 1..4 gate precompute as fp32 WMMA GEMM:
//   pre[dir] = Y[T,256] x Wih[dir]^T[256,512] + bias
// Each wave computes a 16(M=t) x 64(N=j) strip with 4 accumulators,
// K stepped by 4 using V_WMMA_F32_16X16X4_F32 (fp32-faithful to reference).
// ---------------------------------------------------------------------------
__global__ void lstm_gates_wmma(const float* __restrict__ Y,    // [T][256]
                                const float* __restrict__ Wih,  // [2][512][256] layer base
                                const float* __restrict__ bih,  // [2][512] layer base
                                const float* __restrict__ bhh,  // [2][512]
                                float* __restrict__ pre)        // [2][T][512]
{
    int lane = threadIdx.x & 31;
    int wave = blockIdx.x * (blockDim.x >> 5) + (threadIdx.x >> 5);
    // waves: dir(2) x mt(1024) x ns(8)
    int dir = wave >> 13;
    int rem = wave & 8191;
    int mt  = rem >> 3;          // M tile (16 rows of t)
    int ns  = rem & 7;           // N strip of 64 columns (j)

    int m  = lane & 15;
    int kh = lane >> 4;          // 0 or 1

    const float* Yrow  = Y + (size_t)(mt * 16 + m) * 256;
    const float* Wbase = Wih + (size_t)dir * 512 * 256;

    v8f acc[4] = {};

    for (int kk = 0; kk < 256; kk += 4) {
        v2f a = *(const v2f*)(Yrow + kk + 2 * kh);
        #pragma unroll
        for (int tn = 0; tn < 4; ++tn) {
            int j = ns * 64 + tn * 16 + m;
            v2f b = *(const v2f*)(Wbase + (size_t)j * 256 + kk + 2 * kh);
            acc[tn] = __builtin_amdgcn_wmma_f32_16x16x4_f32(
                false, a, false, b, (short)0, acc[tn], false, false);
        }
    }

    float* preD = pre + (size_t)dir * T * 512;
    #pragma unroll
    for (int tn = 0; tn < 4; ++tn) {
        int j = ns * 64 + tn * 16 + m;
        float bias = bih[dir * 512 + j] + bhh[dir * 512 + j];
        #pragma unroll
        for (int v = 0; v < 8; ++v) {
            int trow = mt * 16 + v + 8 * kh;
            preD[(size_t)trow * 512 + j] = acc[tn][v] + bias;
        }
    }
}

// ---------------------------------------------------------------------------
// Sequential recurrence, one workgroup per direction (grid = 2).
// 512 threads: thread j owns W_hh row j as 64 float2 register pairs.
// Dot product: 4 independent float2 chains (ILP) via elementwise fma
// (-> v_pk_fma_f32, 2 MACs/lane/instr); gate pre-activation from global
// memory is added at the END so its load overlaps the whole dot.
// ---------------------------------------------------------------------------
__global__ void lstm_recur(const float* __restrict__ pre,  // [2][T][512]
                           const float* __restrict__ whh,  // [2][512][128] layer base
                           const float* __restrict__ h0,   // [2][128] layer base
                           const float* __restrict__ c0,   // [2][128]
                           float* __restrict__ Yout)       // [T][256]
{
    int dir = blockIdx.x;
    int j   = threadIdx.x;                      // 0..511

    __shared__ __align__(16) float sh[H];
    __shared__ __align__(16) float sc[H];
    __shared__ __align__(16) float sg[512];

    // Register-resident W_hh row, stored as 64 packed float2.
    v2f w2[H / 2];
    const float* wrow = whh + ((size_t)dir * 512 + j) * H;
    #pragma unroll
    for (int k = 0; k < H / 2; ++k) w2[k] = *(const v2f*)(wrow + 2 * k);

    if (j < H) { sh[j] = h0[dir * H + j]; sc[j] = c0[dir * H + j]; }
    __syncthreads();

    const float* preD = pre + (size_t)dir * T * 512;

    for (int s = 0; s < T; ++s) {
        int t  = dir ? (T - 1 - s) : s;
        int tn = dir ? (t - 1) : (t + 1);
        if (s + 1 < T) __builtin_prefetch(preD + (size_t)tn * 512 + j, 0, 1);

        // Issue the gate-preactivation load now; consume it after the dot.
        float preval = preD[(size_t)t * 512 + j];

        v2f d0 = {0.0f, 0.0f}, d1 = d0, d2 = d0, d3 = d0;
        #pragma unroll
        for (int k = 0; k < 32; k += 4) {
            float4 h0v = *(const float4*)(sh + k);          // LDS broadcast reads
            float4 h1v = *(const float4*)(sh + 32 + k);
            float4 h2v = *(const float4*)(sh + 64 + k);
            float4 h3v = *(const float4*)(sh + 96 + k);
            int p = k >> 1;
            d0 = __builtin_elementwise_fma(w2[p +  0], (v2f){h0v.x, h0v.y}, d0);
            d0 = __builtin_elementwise_fma(w2[p +  1], (v2f){h0v.z, h0v.w}, d0);
            d1 = __builtin_elementwise_fma(w2[p + 16], (v2f){h1v.x, h1v.y}, d1);
            d1 = __builtin_elementwise_fma(w2[p + 17], (v2f){h1v.z, h1v.w}, d1);
            d2 = __builtin_elementwise_fma(w2[p + 32], (v2f){h2v.x, h2v.y}, d2);
            d2 = __builtin_elementwise_fma(w2[p + 33], (v2f){h2v.z, h2v.w}, d2);
            d3 = __builtin_elementwise_fma(w2[p + 48], (v2f){h3v.x, h3v.y}, d3);
            d3 = __builtin_elementwise_fma(w2[p + 49], (v2f){h3v.z, h3v.w}, d3);
        }
        v2f dd = (d0 + d1) + (d2 + d3);
        sg[j] = dd.x + dd.y + preval;
        __syncthreads();

        if (j < H) {
            float ig = sg[j];
            float fg = sg[H + j];
            float gg = sg[2 * H + j];
            float og = sg[3 * H + j];
            float c  = sigm(fg) * sc[j] + sigm(ig) * tanh_fast(gg);
            float h  = sigm(og) * tanh_fast(c);
            sc[j] = c;
            sh[j] = h;
            Yout[(size_t)t * 256 + dir * H + j] = h;
        }
        __syncthreads();
    }
}

// ---------------------------------------------------------------------------
// BN (eval mode, running stats) + FC head on the last timestep.
// ---------------------------------------------------------------------------
__global__ void lstm_head(const float* __restrict__ Y,      // [T][256]
                          const float* __restrict__ gamma,
                          const float* __restrict__ beta,
                          const float* __restrict__ mean,
                          const float* __restrict__ var,
                          const float* __restrict__ fcw,    // [2][256]
                          const float* __restrict__ fcb,    // [2]
                          float* __restrict__ out)          // [2]
{
    __shared__ float bn[256];
    int j = threadIdx.x;
    float v = Y[(size_t)(T - 1) * 256 + j];
    bn[j] = (v - mean[j]) * rsqrtf(var[j] + 1e-5f) * gamma[j] + beta[j];
    __syncthreads();
    if (j < 2) {
        float s = fcb[j];
        for (int k = 0; k < 256; ++k) s += bn[k] * fcw[j * 256 + k];
        out[j] = s;
    }
}

// ---------------------------------------------------------------------------
extern "C" void kernel_launch(void* const* d_in, const int* in_sizes, int n_in,
                              void* d_out, int out_size, void* d_ws, size_t ws_size,
                              hipStream_t stream) {
    const float* x       = (const float*)d_in[0];   // [1,T,6]
    const float* h0      = (const float*)d_in[1];   // [10,1,128]
    const float* c0      = (const float*)d_in[2];   // [10,1,128]
    const float* w_ih_l0 = (const float*)d_in[3];   // [2,512,6]
    const float* w_ih    = (const float*)d_in[4];   // [4,2,512,256]
    const float* w_hh    = (const float*)d_in[5];   // [5,2,512,128]
    const float* b_ih    = (const float*)d_in[6];   // [5,2,512]
    const float* b_hh    = (const float*)d_in[7];   // [5,2,512]
    const float* gamma   = (const float*)d_in[8];
    const float* beta    = (const float*)d_in[9];
    const float* mean    = (const float*)d_in[10];
    const float* var     = (const float*)d_in[11];
    const float* fcw     = (const float*)d_in[12];
    const float* fcb     = (const float*)d_in[13];

    // Workspace layout (all fp32, L2-resident):
    //   pre  : [2][T][512]  = 64 MB
    //   buf0 : [T][256]     = 16 MB
    //   buf1 : [T][256]     = 16 MB
    float* pre  = (float*)d_ws;
    float* buf0 = pre + (size_t)2 * T * 512;
    float* buf1 = buf0 + (size_t)T * 256;

    const float* Yin = nullptr;
    for (int l = 0; l < LAYERS; ++l) {
        if (l == 0) {
            int total = 2 * T * 512;
            lstm_pre_l0<<<total / 256, 256, 0, stream>>>(
                x, w_ih_l0, b_ih, b_hh, pre);
        } else {
            // 16384 waves of work -> 2048 blocks x 256 threads (8 waves)
            lstm_gates_wmma<<<2048, 256, 0, stream>>>(
                Yin,
                w_ih + (size_t)(l - 1) * 2 * 512 * 256,
                b_ih + (size_t)l * 2 * 512,
                b_hh + (size_t)l * 2 * 512,
                pre);
        }
        float* Yout = (l & 1) ? buf1 : buf0;
        lstm_recur<<<2, 512, 0, stream>>>(
            pre,
            w_hh + (size_t)l * 2 * 512 * H,
            h0 + (size_t)2 * l * H,
            c0 + (size_t)2 * l * H,
            Yout);
        Yin = Yout;
    }

    lstm_head<<<1, 256, 0, stream>>>(Yin, gamma, beta, mean, var, fcw, fcb,
                                     (float*)d_out);
}